// FlowPredictor_42125039239963
// MI455X (gfx1250) — compile-verified
//
#include <hip/hip_runtime.h>
#include <hip/hip_bf16.h>

#define HID 128

typedef float v2f __attribute__((ext_vector_type(2)));
typedef float v8f __attribute__((ext_vector_type(8)));

// D = A(16x4 f32) * B(4x16 f32) + C, full-rate fp32 WMMA on gfx1250.
__device__ __forceinline__ v8f wmma4(v2f a, v2f b, v8f c) {
    return __builtin_amdgcn_wmma_f32_16x16x4_f32(
        /*neg_a=*/false, a, /*neg_b=*/false, b,
        /*c_mod=*/(short)0, c, /*reuse_a=*/false, /*reuse_b=*/false);
}

__global__ void zero_kernel(float* __restrict__ p, int n) {
    int i = blockIdx.x * blockDim.x + threadIdx.x;
    if (i < n) p[i] = 0.0f;
}

// h[n,k] = sum_{j<5} x[n,j] * enc_w[j,k] + enc_b[k]
__global__ void enc_kernel(const float* __restrict__ x, const float* __restrict__ w,
                           const float* __restrict__ b, float* __restrict__ h, int N) {
    int n = blockIdx.x;
    int k = threadIdx.x;
    float acc = b[k];
#pragma unroll
    for (int j = 0; j < 5; ++j) acc += x[n * 5 + j] * w[j * HID + k];
    h[(size_t)n * HID + k] = acc;
}

// Y[N,128] = concat_K(X0,[X1]) @ W[K,128]  (+ bias (optionally scaled per-row))
//            (+ residual) (optional relu).  K = 128 (X1==null) or 256.
// One 16x16 output tile per wave; 8 waves/block cover 16 rows x 128 cols.
// W is staged 128x128 at a time in 64 KB LDS shared by the whole block.
__global__ __launch_bounds__(256) void gemm_kernel(
    const float* __restrict__ X0, const float* __restrict__ X1,
    const float* __restrict__ W, const float* __restrict__ bias,
    const float* __restrict__ rowScale, const float* __restrict__ res,
    float* __restrict__ Y, int nrows, int relu) {
    __shared__ float sW[HID * HID];  // 64 KB

    const int lane = threadIdx.x & 31;
    const int wave = threadIdx.x >> 5;
    const int rowbase = blockIdx.x * 16;
    const int colbase = wave * 16;
    const int m = lane & 15;   // A row / B-D column within tile
    const int kh = lane >> 4;  // K-half select (A/B), row-half select (D)
    const int col = colbase + m;

    v8f acc = {0.f, 0.f, 0.f, 0.f, 0.f, 0.f, 0.f, 0.f};

    const int nphase = (X1 != nullptr) ? 2 : 1;
    for (int p = 0; p < nphase; ++p) {
        __syncthreads();
        {   // stage one 128x128 slab of W into LDS (float4, fully coalesced)
            const float4* Wp = (const float4*)(W + (size_t)p * HID * HID);
            float4* s4 = (float4*)sW;
            for (int i = threadIdx.x; i < HID * HID / 4; i += 256) s4[i] = Wp[i];
        }
        __syncthreads();

        const float* Xp = p ? X1 : X0;
        int r = rowbase + m;
        if (r >= nrows) r = nrows - 1;  // clamp (pad rows read garbage, never stored)
        const float* xrow = Xp + (size_t)r * HID + 2 * kh;

#pragma unroll
        for (int k0 = 0; k0 < HID; k0 += 4) {
            // A 16x4: lane(m,kh) holds X[m, k0+2*kh .. +1]   (ISA 7.12.2)
            v2f a = *(const v2f*)(xrow + k0);
            // B 4x16: VGPR v holds K = v + 2*kh, column = lane&15
            v2f b;
            b.x = sW[(k0 + 2 * kh + 0) * HID + col];
            b.y = sW[(k0 + 2 * kh + 1) * HID + col];
            acc = wmma4(a, b, acc);
        }
    }

    // Epilogue: D VGPR v -> row = rowbase + v + 8*kh, col = colbase + (lane&15)
    float bterm = bias ? bias[col] : 0.0f;
#pragma unroll
    for (int v = 0; v < 8; ++v) {
        int row = rowbase + 8 * kh + v;
        if (row >= nrows) continue;
        float val = acc[v];
        float bt = bterm;
        if (rowScale) bt *= rowScale[row];  // degree-scaled bias for aggr GEMM
        val += bt;
        if (res) val += res[(size_t)row * HID + col];  // residual h + update
        if (relu) val = fmaxf(val, 0.0f);
        Y[(size_t)row * HID + col] = val;
    }
}

// One wave per edge.  pre = Pd[dst] + Ps[src] + edge_attr @ W1e  (mb1 folded
// into Pd's bias).  Accumulate relu(pre) into R[dst] and degree into cnt[dst].
__global__ __launch_bounds__(256) void edge_kernel(
    const int* __restrict__ ei, const float* __restrict__ ea,
    const float* __restrict__ Pd, const float* __restrict__ Ps,
    const float* __restrict__ We,  // [3,128]
    float* __restrict__ R, float* __restrict__ cnt, int E) {
    int wid = (blockIdx.x * blockDim.x + threadIdx.x) >> 5;
    int lane = threadIdx.x & 31;
    if (wid >= E) return;
    int src = ei[wid];
    int dst = ei[E + wid];
    float e0 = ea[wid * 3 + 0], e1 = ea[wid * 3 + 1], e2 = ea[wid * 3 + 2];
    int k = lane * 4;
    const float4 pd = *(const float4*)(Pd + (size_t)dst * HID + k);
    const float4 ps = *(const float4*)(Ps + (size_t)src * HID + k);
    const float4 w0 = *(const float4*)(We + 0 * HID + k);
    const float4 w1 = *(const float4*)(We + 1 * HID + k);
    const float4 w2 = *(const float4*)(We + 2 * HID + k);
    float4 pre;
    pre.x = pd.x + ps.x + e0 * w0.x + e1 * w1.x + e2 * w2.x;
    pre.y = pd.y + ps.y + e0 * w0.y + e1 * w1.y + e2 * w2.y;
    pre.z = pd.z + ps.z + e0 * w0.z + e1 * w1.z + e2 * w2.z;
    pre.w = pd.w + ps.w + e0 * w0.w + e1 * w1.w + e2 * w2.w;
    float* rp = R + (size_t)dst * HID + k;
    atomicAdd(rp + 0, fmaxf(pre.x, 0.0f));   // global_atomic_add_f32 @ L2
    atomicAdd(rp + 1, fmaxf(pre.y, 0.0f));
    atomicAdd(rp + 2, fmaxf(pre.z, 0.0f));
    atomicAdd(rp + 3, fmaxf(pre.w, 0.0f));
    if (lane == 0) atomicAdd(cnt + dst, 1.0f);
}

// out[n, j<5] = h2[n,:] @ dec_w + dec_b
__global__ void dec_kernel(const float* __restrict__ h2, const float* __restrict__ w,
                           const float* __restrict__ b, float* __restrict__ out, int N) {
    __shared__ float red[HID];
    int n = blockIdx.x;
    int k = threadIdx.x;
    float v = h2[(size_t)n * HID + k];
    for (int j = 0; j < 5; ++j) {
        red[k] = v * w[k * 5 + j];
        __syncthreads();
        for (int s = 64; s > 0; s >>= 1) {
            if (k < s) red[k] += red[k + s];
            __syncthreads();
        }
        if (k == 0) out[n * 5 + j] = red[0] + b[j];
        __syncthreads();
    }
}

extern "C" void kernel_launch(void* const* d_in, const int* in_sizes, int n_in,
                              void* d_out, int out_size, void* d_ws, size_t ws_size,
                              hipStream_t stream) {
    (void)n_in; (void)out_size; (void)ws_size;
    const float* x      = (const float*)d_in[0];
    const int*   ei     = (const int*)d_in[1];
    const float* ea     = (const float*)d_in[2];
    const float* enc_w  = (const float*)d_in[3];
    const float* enc_b  = (const float*)d_in[4];
    const float* dec_w  = (const float*)d_in[5];
    const float* dec_b  = (const float*)d_in[6];
    const float* msg_w1 = (const float*)d_in[7];
    const float* msg_b1 = (const float*)d_in[8];
    const float* msg_w2 = (const float*)d_in[9];
    const float* msg_b2 = (const float*)d_in[10];
    const float* upd_w1 = (const float*)d_in[11];
    const float* upd_b1 = (const float*)d_in[12];
    const float* upd_w2 = (const float*)d_in[13];
    const float* upd_b2 = (const float*)d_in[14];
    float* out = (float*)d_out;

    const int N = in_sizes[0] / 5;
    const int E = in_sizes[1] / 2;
    const int L = 3;  // only the last layer's h_update survives in the reference

    const float* mw1 = msg_w1 + (size_t)L * (2 * HID + 3) * HID;  // [259,128]
    const float* mb1 = msg_b1 + (size_t)L * HID;
    const float* mw2 = msg_w2 + (size_t)L * HID * HID;
    const float* mb2 = msg_b2 + (size_t)L * HID;
    const float* uw1 = upd_w1 + (size_t)L * 2 * HID * HID;        // [256,128]
    const float* ub1 = upd_b1 + (size_t)L * HID;
    const float* uw2 = upd_w2 + (size_t)L * HID * HID;
    const float* ub2 = upd_b2 + (size_t)L * HID;

    float* ws = (float*)d_ws;
    const size_t NH = (size_t)N * HID;
    float* h   = ws;            // [N,128] node features (live to the end)
    float* Pd  = ws + NH;       // h @ mw1[0:128]  + mb1 ; later reused as aggr
    float* Ps  = ws + 2 * NH;   // h @ mw1[128:256]      ; later reused as hu1
    float* R   = ws + 3 * NH;   // segment_sum(relu(pre)); later reused as h2
    float* cnt = ws + 4 * NH;   // [N] degree (float)
    float* aggr = Pd;
    float* hu1  = Ps;
    float* h2   = R;

    const int rowBlocks = (N + 15) / 16;

    // R and cnt are contiguous: zero both (replays don't re-poison ws).
    {
        int tot = (int)NH + N;
        zero_kernel<<<(tot + 255) / 256, 256, 0, stream>>>(R, tot);
    }
    enc_kernel<<<N, HID, 0, stream>>>(x, enc_w, enc_b, h, N);
    // Pd = h @ W1a + mb1 ; Ps = h @ W1b
    gemm_kernel<<<rowBlocks, 256, 0, stream>>>(h, nullptr, mw1, mb1, nullptr,
                                               nullptr, Pd, N, 0);
    gemm_kernel<<<rowBlocks, 256, 0, stream>>>(h, nullptr, mw1 + HID * HID, nullptr,
                                               nullptr, nullptr, Ps, N, 0);
    // Edge phase: gather, relu, scatter-add (L2-resident).
    edge_kernel<<<(E * 32 + 255) / 256, 256, 0, stream>>>(
        ei, ea, Pd, Ps, mw1 + 2 * HID * HID, R, cnt, E);
    // aggr = R @ mw2 + cnt (x) mb2
    gemm_kernel<<<rowBlocks, 256, 0, stream>>>(R, nullptr, mw2, mb2, cnt,
                                               nullptr, aggr, N, 0);
    // hu1 = relu([h, aggr] @ uw1 + ub1)   (K = 256, two W phases)
    gemm_kernel<<<rowBlocks, 256, 0, stream>>>(h, aggr, uw1, ub1, nullptr,
                                               nullptr, hu1, N, 1);
    // h2 = hu1 @ uw2 + ub2 + h
    gemm_kernel<<<rowBlocks, 256, 0, stream>>>(hu1, nullptr, uw2, ub2, nullptr,
                                               h, h2, N, 0);
    dec_kernel<<<N, HID, 0, stream>>>(h2, dec_w, dec_b, out, N);
}